// PfaffianAuxilliaryFermions_14766097563681
// MI455X (gfx1250) — compile-verified
//
#include <hip/hip_runtime.h>
#include <hip/hip_bf16.h>
#include <math.h>

// ---------------------------------------------------------------------------
// Pfaffian of A = [[F[idx,idx], U[idx]], [-U[idx]^T, 0]]  (n = 1152, fp32)
// via Wimmer Householder tridiagonalization.
//   - A stays resident in L2 (5.3 MB << 192 MB).
//   - rank-2 trailing update  A += v w^T - w v^T  is cast as a K=4 fp32 WMMA
//     GEMM (V_WMMA_F32_16X16X4_F32), K-cols 2,3 zero-padded. Since v,w are
//     zero below the pivot, unmasked full-tile updates are exact.
// ---------------------------------------------------------------------------

#define TWO_N   4096
#define NPART   1024
#define NFREE_  128
#define NDIM    1152   // NPART + NFREE_

typedef __attribute__((ext_vector_type(2))) float v2f;
typedef __attribute__((ext_vector_type(8))) float v8f;

// ---------------- idx = flatnonzero(x > 0), sorted ascending ---------------
__global__ __launch_bounds__(256) void pfaf_idx_kernel(const float* __restrict__ x,
                                                       int* __restrict__ idx)
{
    __shared__ int counts[256];
    __shared__ int offs[256];
    const int t = threadIdx.x;
    const int per = TWO_N / 256;          // 16
    const int base = t * per;
    int c = 0;
    for (int k = 0; k < per; ++k) c += (x[base + k] > 0.f) ? 1 : 0;
    counts[t] = c;
    __syncthreads();
    if (t == 0) {
        int run = 0;
        for (int k = 0; k < 256; ++k) { offs[k] = run; run += counts[k]; }
    }
    __syncthreads();
    int o = offs[t];
    for (int k = 0; k < per; ++k)
        if (x[base + k] > 0.f) idx[o++] = base + k;
}

// ---------------- build A (1152x1152) from F, U, idx -----------------------
__global__ __launch_bounds__(256) void pfaf_build_kernel(const float* __restrict__ F,
                                                         const float* __restrict__ U,
                                                         const int* __restrict__ idx,
                                                         float* __restrict__ A)
{
    const int tid = blockIdx.x * 256 + threadIdx.x;
    if (tid >= NDIM * NDIM) return;
    const int r = tid / NDIM;
    const int c = tid - r * NDIM;
    float val = 0.f;
    if (r < NPART && c < NPART) {
        const int ig = idx[r], jg = idx[c];
        if (ig > jg)      val =  F[(size_t)ig * (ig - 1) / 2 + jg];
        else if (ig < jg) val = -F[(size_t)jg * (jg - 1) / 2 + ig];
    } else if (r < NPART) {                 // top-right:  U[idx[r], c-NPART]
        val = U[(size_t)idx[r] * NFREE_ + (c - NPART)];
    } else if (c < NPART) {                 // bottom-left: -U[idx[c], r-NPART]
        val = -U[(size_t)idx[c] * NFREE_ + (r - NPART)];
    }
    A[(size_t)r * NDIM + c] = val;
}

// ---------------- scal[0]=tau, scal[1]=pf ----------------------------------
__global__ void pfaf_init_kernel(float* __restrict__ scal)
{
    scal[0] = 0.f;
    scal[1] = 1.f;
}

// ---------------- per-step reflector: build v, tau; fold pf ----------------
__global__ __launch_bounds__(256) void pfaf_householder_kernel(const float* __restrict__ A,
                                                               float* __restrict__ v,
                                                               float* __restrict__ scal,
                                                               int i)
{
    __shared__ float red[256];
    __shared__ float sh[2];
    const int t = threadIdx.x;
    const float x0 = A[(size_t)(i + 1) * NDIM + i];
    float s2 = 0.f;
    for (int j = i + 2 + t; j < NDIM; j += 256) {
        const float c = A[(size_t)j * NDIM + i];
        s2 += c * c;
    }
    red[t] = s2;
    __syncthreads();
    for (int stride = 128; stride > 0; stride >>= 1) {
        if (t < stride) red[t] += red[t + stride];
        __syncthreads();
    }
    if (t == 0) {
        const float sigma  = red[0];
        const float norm_x = sqrtf(x0 * x0 + sigma);
        const float s      = (x0 > 0.f) ? 1.f : -1.f;
        const float head   = x0 + s * norm_x;
        const float vnorm  = sqrtf(head * head + sigma);
        const bool  nz     = sigma > 0.f;
        const float inv    = nz ? (1.f / ((vnorm == 0.f) ? 1.f : vnorm)) : 0.f;
        const float tau    = nz ? 2.f : 0.f;
        const float alpha  = nz ? (-s * norm_x) : x0;
        scal[0] = tau;
        scal[1] = scal[1] * (1.f - tau) * (((i & 1) == 0) ? -alpha : 1.f);
        sh[0] = inv;
        sh[1] = head;
    }
    __syncthreads();
    const float inv = sh[0], head = sh[1];
    for (int j = t; j < NDIM; j += 256) {
        float val;
        if (j <= i)           val = 0.f;
        else if (j == i + 1)  val = head * inv;
        else                  val = A[(size_t)j * NDIM + i] * inv;
        v[j] = val;
    }
}

// ---------------- w = tau * (A @ v), masked to rows > i --------------------
__global__ __launch_bounds__(256) void pfaf_matvec_kernel(const float* __restrict__ A,
                                                          const float* __restrict__ v,
                                                          float* __restrict__ w,
                                                          const float* __restrict__ scal,
                                                          int i)
{
    const int wid  = threadIdx.x >> 5;
    const int lane = threadIdx.x & 31;
    const int r    = blockIdx.x * 8 + wid;
    if (r >= NDIM) return;
    if (r <= i) { if (lane == 0) w[r] = 0.f; return; }
    const float tau = scal[0];
    const float* Ar = A + (size_t)r * NDIM;
    float acc = 0.f;
    for (int j = i + 1 + lane; j < NDIM; j += 32) acc += Ar[j] * v[j];
    #pragma unroll
    for (int off = 16; off > 0; off >>= 1) acc += __shfl_down(acc, off, 32);
    if (lane == 0) w[r] = tau * acc;
}

// ---------------- A += v w^T - w v^T via fp32 WMMA 16x16x4 -----------------
// One wave per 16x16 tile. A-op (16x4): K0=v, K1=w, K2=K3=0 (lanes 16-31
// carry K2/K3 per ISA layout -> zeros). B-op (4x16): K0=w, K1=-v, rest 0.
// C/D layout: VGPR j -> M = j (+8 for lanes 16-31), N = lane&15.
__global__ __launch_bounds__(32) void pfaf_rank2_wmma_kernel(float* __restrict__ A,
                                                             const float* __restrict__ v,
                                                             const float* __restrict__ w,
                                                             int t0, int ntiles)
{
    const int tile = blockIdx.x;
    const int tr = tile / ntiles;
    const int tc = tile - tr * ntiles;
    const int row0 = t0 + tr * 16;
    const int col0 = t0 + tc * 16;
    const int lane = threadIdx.x;          // 0..31, full wave
    const int l15  = lane & 15;
    const bool lo  = lane < 16;

    // Branchless operand builds (EXEC must be all-ones at the WMMA).
    const float vr = v[row0 + l15];
    const float wr = w[row0 + l15];
    v2f a;
    a.x = lo ? vr : 0.f;                   // K=0 column of A-op
    a.y = lo ? wr : 0.f;                   // K=1 column of A-op
    const float wc = w[col0 + l15];
    const float vc = v[col0 + l15];
    v2f b;
    b.x = lo ? wc : 0.f;                   // K=0 row of B-op
    b.y = lo ? -vc : 0.f;                  // K=1 row of B-op

    const int mhi = (lane >> 4) * 8;
    const int cc  = col0 + l15;
    v8f cacc;
    #pragma unroll
    for (int j = 0; j < 8; ++j)
        cacc[j] = A[(size_t)(row0 + j + mhi) * NDIM + cc];

    cacc = __builtin_amdgcn_wmma_f32_16x16x4_f32(
        /*neg_a=*/false, a, /*neg_b=*/false, b,
        /*c_mod=*/(short)0, cacc, /*reuse_a=*/false, /*reuse_b=*/false);

    #pragma unroll
    for (int j = 0; j < 8; ++j)
        A[(size_t)(row0 + j + mhi) * NDIM + cc] = cacc[j];
}

// ---------------- out = pf * A[n-2, n-1] -----------------------------------
__global__ void pfaf_final_kernel(const float* __restrict__ A,
                                  const float* __restrict__ scal,
                                  float* __restrict__ out)
{
    out[0] = scal[1] * A[(size_t)(NDIM - 2) * NDIM + (NDIM - 1)];
}

// ---------------------------------------------------------------------------
extern "C" void kernel_launch(void* const* d_in, const int* in_sizes, int n_in,
                              void* d_out, int out_size, void* d_ws, size_t ws_size,
                              hipStream_t stream)
{
    const float* x = (const float*)d_in[0];
    const float* F = (const float*)d_in[1];
    const float* U = (const float*)d_in[2];
    float* out = (float*)d_out;

    // Workspace layout (fp32 words): A | v | w | scal(4) | idx(int)
    float* A    = (float*)d_ws;
    float* v    = A + (size_t)NDIM * NDIM;
    float* w    = v + NDIM;
    float* scal = w + NDIM;
    int*   idx  = (int*)(scal + 4);

    pfaf_idx_kernel<<<1, 256, 0, stream>>>(x, idx);
    pfaf_init_kernel<<<1, 1, 0, stream>>>(scal);
    {
        const int total = NDIM * NDIM;
        pfaf_build_kernel<<<(total + 255) / 256, 256, 0, stream>>>(F, U, idx, A);
    }

    for (int i = 0; i < NDIM - 2; ++i) {
        pfaf_householder_kernel<<<1, 256, 0, stream>>>(A, v, scal, i);
        pfaf_matvec_kernel<<<(NDIM + 7) / 8, 256, 0, stream>>>(A, v, w, scal, i);
        const int t0     = ((i + 1) / 16) * 16;    // first tile row/col touched
        const int ntiles = (NDIM - t0) / 16;       // NDIM is a multiple of 16
        pfaf_rank2_wmma_kernel<<<ntiles * ntiles, 32, 0, stream>>>(A, v, w, t0, ntiles);
    }

    pfaf_final_kernel<<<1, 1, 0, stream>>>(A, scal, out);
}